// LLClusterCoordinates_618475290650
// MI455X (gfx1250) — compile-verified
//
#include <hip/hip_runtime.h>

// Object-condensation clustering loss (beta=0.5, Q_MIN=1, S_B=0, avg-cc-pos)
// for MI455X / gfx1250. Distance-matrix core uses V_WMMA_F32_16X16X4_F32.
// Round 3: 3-phase postprocess so the 8 independent v_sqrt_f32 issue
// back-to-back and fill each other's TRANS hazard slots (kills the v_nops).

#define K_PER_EVENT 500
#define KPAD        512     // clusters padded per event for 16-wide tiling
#define NEV         2

#define ATANH_HALF 0.5493061443340548f
#define QCONST (ATANH_HALF * ATANH_HALF + 1.0f)   // arctanh(0.5)^2 + Q_MIN

typedef __attribute__((ext_vector_type(2))) float v2f;
typedef __attribute__((ext_vector_type(8))) float v8f;

// ---------------- Phase A: zero accumulators ----------------
__global__ void oc_zero(float* __restrict__ csum, float* __restrict__ out) {
  int i = blockIdx.x * blockDim.x + threadIdx.x;
  if (i < NEV * KPAD * 4) csum[i] = 0.0f;
  if (i == 0) out[0] = 0.0f;
}

// ---------------- Phase B: per-cluster coord sums + counts ----------------
__global__ void oc_accum(const float* __restrict__ coords,
                         const int* __restrict__ truth,
                         const int* __restrict__ rs,
                         float* __restrict__ csum, int N) {
  int i = blockIdx.x * blockDim.x + threadIdx.x;
  if (i >= N) return;
  int ev = (i >= rs[1]) ? 1 : 0;                 // 2 events (row_splits sorted)
  int k  = ev * KPAD + truth[i];
  float x = coords[3 * i + 0];
  float y = coords[3 * i + 1];
  float z = coords[3 * i + 2];
  atomicAdd(&csum[4 * k + 0], x);
  atomicAdd(&csum[4 * k + 1], y);
  atomicAdd(&csum[4 * k + 2], z);
  atomicAdd(&csum[4 * k + 3], 1.0f);
}

// ---------------- Phase C: finalize centers, c^2, q_alpha ----------------
__global__ void oc_centers(const float* __restrict__ csum,
                           float4* __restrict__ centers,
                           float* __restrict__ qa) {
  int k = blockIdx.x * blockDim.x + threadIdx.x;
  if (k >= NEV * KPAD) return;
  float sx  = csum[4 * k + 0];
  float sy  = csum[4 * k + 1];
  float sz  = csum[4 * k + 2];
  float cnt = csum[4 * k + 3];
  bool alive = ((k & (KPAD - 1)) < K_PER_EVENT) && (cnt > 0.0f);
  float sumq = QCONST * cnt;                      // q is constant per hit
  float inv  = 1.0f / fmaxf(sumq, 1e-6f);
  float cx = alive ? QCONST * sx * inv : 0.0f;
  float cy = alive ? QCONST * sy * inv : 0.0f;
  float cz = alive ? QCONST * sz * inv : 0.0f;
  centers[k] = make_float4(cx, cy, cz, cx * cx + cy * cy + cz * cz);
  qa[k] = alive ? QCONST : 0.0f;
}

// ---------------- Phase D: WMMA distance tiles + loss reduction ----------------
// One wave32 owns a 16-point tile; sweeps 32 cluster tiles (512 padded
// clusters) of its event. D = A(-2*coords, 16x4) x B(centers, 4x16) + C(x2+c2)
// gives d^2 directly via V_WMMA_F32_16X16X4_F32.
__global__ void __launch_bounds__(256)
oc_loss(const float* __restrict__ coords,
        const int* __restrict__ truth,
        const int* __restrict__ rs,
        const float4* __restrict__ centers,
        const float* __restrict__ qa,
        float* __restrict__ out, int N) {
  const int lane = threadIdx.x & 31;
  const int half = lane >> 4;                     // 0: lanes 0-15, 1: lanes 16-31
  const int wid  = blockIdx.x * (blockDim.x >> 5) + (threadIdx.x >> 5);
  const int base = wid * 16;
  if (base >= N) return;                          // wave-uniform exit

  const int r0 = rs[0], r1 = rs[1], r2 = rs[2];
  const int ev    = (base >= r1) ? 1 : 0;
  const int evBeg = ev ? r1 : r0;
  const int evEnd = ev ? r2 : r1;
  const float nev = (float)(evEnd - evBeg);

  // ---- A fragment (ISA 32-bit 16x4 layout):
  //   VGPR0: lanes 0-15 K=0 (x), lanes 16-31 K=2 (z)
  //   VGPR1: lanes 0-15 K=1 (y), lanes 16-31 K=3 (pad 0)
  // pre-scaled by -2 so D = x^2 - 2 x.c + c^2.
  const int gA = base + (lane & 15);
  const bool actA = (gA < N) && (gA >= evBeg) && (gA < evEnd);
  float ax = 0.0f, ay = 0.0f, az = 0.0f;
  if (actA) {
    ax = coords[3 * gA + 0];
    ay = coords[3 * gA + 1];
    az = coords[3 * gA + 2];
  }
  v2f A;
  A.x = half ? (-2.0f * az) : (-2.0f * ax);
  A.y = half ? 0.0f : (-2.0f * ay);

  // ---- per-accumulator-register point metadata (C/D layout: M = r + 8*half)
  float x2h[8];
  int   th[8];
#pragma unroll
  for (int r = 0; r < 8; ++r) {
    int g = base + r + 8 * half;
    bool act = (g < N) && (g >= evBeg) && (g < evEnd);
    if (act) {
      float x = coords[3 * g + 0];
      float y = coords[3 * g + 1];
      float z = coords[3 * g + 2];
      x2h[r] = x * x + y * y + z * z;
      th[r]  = truth[g];
    } else {
      x2h[r] = 1e30f;   // kills repulsive hinge for inactive rows
      th[r]  = -1;      // never a member
    }
  }

  float acc = 0.0f;
#pragma unroll 4
  for (int ct = 0; ct < KPAD / 16; ++ct) {
    const int n = ct * 16 + (lane & 15);          // cluster column 0..511
    const float4 cb = centers[ev * KPAD + n];     // (cx, cy, cz, c^2)
    const float  qn = qa[ev * KPAD + n];          // 0 for pad / dead clusters

    // B fragment (4x16, K rows striped like A): VGPR0 = K0/K2, VGPR1 = K1/K3
    v2f B;
    B.x = half ? cb.z : cb.x;
    B.y = half ? 0.0f : cb.y;

    v8f C;
#pragma unroll
    for (int r = 0; r < 8; ++r) C[r] = x2h[r] + cb.w;

    // d^2 tile: D[m][n] = x2[m] + c2[n] - 2 * coords[m] . centers[n]
    v8f D = __builtin_amdgcn_wmma_f32_16x16x4_f32(
        /*neg_a=*/false, A, /*neg_b=*/false, B,
        /*c_mod=*/(short)0, C, /*reuse_a=*/false, /*reuse_b=*/false);

    // ---- 3-phase postprocess: keep the 8 v_sqrt_f32 independent so they
    // issue back-to-back (each fills the previous one's TRANS hazard slot).
    float d2[8], sq[8];
#pragma unroll
    for (int r = 0; r < 8; ++r)
      d2[r] = __builtin_amdgcn_fmed3f(D[r], 0.0f, 3.0e38f);  // clamp, 1 op
#pragma unroll
    for (int r = 0; r < 8; ++r)
      sq[r] = __builtin_amdgcn_sqrtf(d2[r] + 1e-6f);         // raw v_sqrt_f32
#pragma unroll
    for (int r = 0; r < 8; ++r) {
      // hinge relu(1 - s): 1 - s <= 1 always, so clamp to [0,1] is exact
      float rep  = __builtin_amdgcn_fmed3f(1.0f - sq[r], 0.0f, 1.0f);
      float term = (th[r] == n) ? d2[r] : rep;    // member ? attractive : hinge
      acc = fmaf(qn, term, acc);
    }
  }

  // ---- wave32 butterfly reduction, one atomic per wave
#pragma unroll
  for (int off = 16; off > 0; off >>= 1) acc += __shfl_xor(acc, off, 32);
  if (lane == 0) {
    // scale by per-hit weight w = q / n_ev and 1/N_EVENTS
    atomicAdd(out, acc * (QCONST / (nev * (float)NEV)));
  }
}

// ---------------- launch ----------------
extern "C" void kernel_launch(void* const* d_in, const int* in_sizes, int n_in,
                              void* d_out, int out_size, void* d_ws, size_t ws_size,
                              hipStream_t stream) {
  const float* coords = (const float*)d_in[0];
  const int*   truth  = (const int*)d_in[1];
  const int*   rs     = (const int*)d_in[2];
  float*       out    = (float*)d_out;

  const int N = in_sizes[0] / 3;

  // workspace layout (36,864 bytes total)
  float*  csum    = (float*)d_ws;                                   // 1024 float4
  float4* centers = (float4*)((char*)d_ws + 4096 * sizeof(float));  // 1024 float4
  float*  qa      = (float*)((char*)d_ws + 4096 * sizeof(float)
                                         + 1024 * sizeof(float4));  // 1024 float

  oc_zero<<<(NEV * KPAD * 4 + 255) / 256, 256, 0, stream>>>(csum, out);
  oc_accum<<<(N + 255) / 256, 256, 0, stream>>>(coords, truth, rs, csum, N);
  oc_centers<<<(NEV * KPAD + 255) / 256, 256, 0, stream>>>(csum, centers, qa);

  const int nwaves = (N + 15) / 16;               // one wave per 16-point tile
  const int blocks = (nwaves + 7) / 8;            // 8 waves (256 threads) / block
  oc_loss<<<blocks, 256, 0, stream>>>(coords, truth, rs, centers, qa, out, N);
}